// MMDLoss_65670049956029
// MI455X (gfx1250) — compile-verified
//
#include <hip/hip_runtime.h>

// ---------------------------------------------------------------------------
// MMD loss (multi-bandwidth Gaussian kernel) for s,t in [8192, 256] f32.
//  - Gram matrices on the fp32 matrix pipe: V_WMMA_F32_16X16X4_F32
//  - k_xx / k_yy exploit symmetry (upper-triangle tiles only, x2 weight)
//  - Double-buffered LDS fed by GLOBAL_LOAD_ASYNC_TO_LDS_B128 (ASYNCcnt)
//  - Deterministic per-WG partials + double-precision finalize
// ---------------------------------------------------------------------------

typedef float v2f __attribute__((ext_vector_type(2)));
typedef float v8f __attribute__((ext_vector_type(8)));

#define NROWS 8192
#define DDIM  256
#define BM    128
#define BN    128
#define BK    16
#define NCHUNK (DDIM / BK)   // 16
#define LDST  20             // padded LDS row stride (floats): 80B = 16B aligned, conflict-free
#define TILES_PER_PAIR (64 * 64)

// Async copy of 16 bytes per lane from the global address gptr to the LDS offset lds_off (tracked by ASYNCcnt).
// The low 32 bits of the flat shared-pointer are the LDS offset (ISA 10.2: LDS aperture
// maps addr[31:0] directly to the wave's LDS offset).
__device__ __forceinline__ void async_copy_b128(unsigned lds_off, const float* gptr) {
    asm volatile("global_load_async_to_lds_b128 %0, %1, off"
                 :: "v"(lds_off), "v"(gptr)
                 : "memory");
}

__global__ __launch_bounds__(256)
void mmd_tile_kernel(const float* __restrict__ X, const float* __restrict__ Y,
                     float* __restrict__ partials, int symmetric)
{
    __shared__ float Xs[2][BM * LDST];
    __shared__ float Ys[2][BN * LDST];
    __shared__ float xn[BM];
    __shared__ float yn[BN];
    __shared__ float red[256];

    const int tid = threadIdx.x;
    const int bm  = blockIdx.x;            // row tile (0..63)
    const int bn  = blockIdx.y;            // col tile (0..63)

    // Symmetric pairs (xx, yy): only upper triangle of tiles; weight x2 off-diag.
    if (symmetric && bm > bn) {
        if (tid == 0) partials[bm * 64 + bn] = 0.0f;
        return;
    }

    const float* Xg0 = X + (size_t)bm * BM * DDIM;
    const float* Yg0 = Y + (size_t)bn * BN * DDIM;

    // Per-thread copy slots: 128 rows x 4 float4 per tile; 2 slots per thread.
    const int row0 = tid >> 2,           c0f = (tid & 3) * 4;
    const int row1 = (tid + 256) >> 2,   c1f = ((tid + 256) & 3) * 4;

    #define ISSUE_CHUNK(kc, buf)                                                        \
        do {                                                                            \
            const int kbase = (kc) * BK;                                                \
            async_copy_b128((unsigned)(size_t)&Xs[buf][row0 * LDST + c0f],              \
                            Xg0 + (size_t)row0 * DDIM + kbase + c0f);                   \
            async_copy_b128((unsigned)(size_t)&Ys[buf][row0 * LDST + c0f],              \
                            Yg0 + (size_t)row0 * DDIM + kbase + c0f);                   \
            async_copy_b128((unsigned)(size_t)&Xs[buf][row1 * LDST + c1f],              \
                            Xg0 + (size_t)row1 * DDIM + kbase + c1f);                   \
            async_copy_b128((unsigned)(size_t)&Ys[buf][row1 * LDST + c1f],              \
                            Yg0 + (size_t)row1 * DDIM + kbase + c1f);                   \
        } while (0)

    // Kick off the first chunk before doing the row norms (overlaps them).
    ISSUE_CHUNK(0, 0);

    // ---- row norms for this tile (inputs are L2-resident; cheap) ----------
    {
        const int r = tid & 127;
        const float* rowp = (tid < 128) ? (X + ((size_t)bm * BM + r) * DDIM)
                                        : (Y + ((size_t)bn * BN + r) * DDIM);
        const float4* rp4 = reinterpret_cast<const float4*>(rowp);
        float s = 0.0f;
        #pragma unroll 8
        for (int i = 0; i < DDIM / 4; ++i) {
            float4 v = rp4[i];
            s += v.x * v.x + v.y * v.y + v.z * v.z + v.w * v.w;
        }
        if (tid < 128) xn[r] = s; else yn[r] = s;
    }

    // ---- wave tiling ------------------------------------------------------
    const int wave = tid >> 5;
    const int lane = tid & 31;
    const int half = lane >> 4;            // 0: lanes 0-15, 1: lanes 16-31
    const int lr   = lane & 15;
    const int m0   = (wave >> 1) * 32;     // 4 x 32-row bands
    const int n0   = (wave & 1) * 64;      // 2 x 64-col bands

    v8f acc[2][4];
    #pragma unroll
    for (int i = 0; i < 2; ++i)
        #pragma unroll
        for (int j = 0; j < 4; ++j)
            acc[i][j] = (v8f){0.f, 0.f, 0.f, 0.f, 0.f, 0.f, 0.f, 0.f};

    // ---- K loop: double-buffered async chunks, 4 WMMA k-steps each --------
    for (int kc = 0; kc < NCHUNK; ++kc) {
        const int cur = kc & 1;
        if (kc + 1 < NCHUNK) {
            ISSUE_CHUNK(kc + 1, cur ^ 1);
            asm volatile("s_wait_asynccnt 4" ::: "memory");  // chunk kc landed (in-order)
        } else {
            asm volatile("s_wait_asynccnt 0" ::: "memory");
        }
        __syncthreads();   // all waves' async writes visible

        #pragma unroll
        for (int ks = 0; ks < BK / 4; ++ks) {
            const int k0 = ks * 4 + 2 * half;   // lanes 0-15 -> K={0,1}, lanes 16-31 -> K={2,3}
            v2f a[2], b[4];
            #pragma unroll
            for (int i = 0; i < 2; ++i)
                a[i] = *reinterpret_cast<const v2f*>(&Xs[cur][(m0 + i * 16 + lr) * LDST + k0]);
            #pragma unroll
            for (int j = 0; j < 4; ++j)
                b[j] = *reinterpret_cast<const v2f*>(&Ys[cur][(n0 + j * 16 + lr) * LDST + k0]);
            #pragma unroll
            for (int i = 0; i < 2; ++i)
                #pragma unroll
                for (int j = 0; j < 4; ++j)
                    acc[i][j] = __builtin_amdgcn_wmma_f32_16x16x4_f32(
                        false, a[i], false, b[j], (short)0, acc[i][j], false, false);
        }
        __syncthreads();   // readers done before next iteration overwrites buf
    }

    // ---- epilogue: dist^2, clamp, 5-bandwidth exp sum ---------------------
    // 1/(2*bw^2) for bw in {0.2, 0.5, 1, 2, 5}
    const float c0 = 12.5f, c1 = 2.0f, c2 = 0.5f, c3 = 0.125f, c4 = 0.02f;
    float local = 0.0f;
    #pragma unroll
    for (int i = 0; i < 2; ++i) {
        #pragma unroll
        for (int j = 0; j < 4; ++j) {
            const float yv = yn[n0 + j * 16 + lr];
            #pragma unroll
            for (int v = 0; v < 8; ++v) {
                const int mrow = m0 + i * 16 + v + 8 * half;  // C/D layout: VGPR v -> M = v + 8*half
                float d = xn[mrow] + yv - 2.0f * acc[i][j][v];
                d = fmaxf(d, 0.0f);
                local += __expf(-d * c0) + __expf(-d * c1) + __expf(-d * c2)
                       + __expf(-d * c3) + __expf(-d * c4);
            }
        }
    }

    // ---- block reduce -> one weighted partial per workgroup ---------------
    red[tid] = local;
    __syncthreads();
    #pragma unroll
    for (int s = 128; s > 0; s >>= 1) {
        if (tid < s) red[tid] += red[tid + s];
        __syncthreads();
    }
    if (tid == 0) {
        const float w = (symmetric && bm != bn) ? 2.0f : 1.0f;
        partials[bm * 64 + bn] = w * red[0];
    }
    #undef ISSUE_CHUNK
}

__global__ __launch_bounds__(256)
void mmd_finalize(const float* __restrict__ partials, float* __restrict__ out)
{
    __shared__ double red[256];
    const int tid = threadIdx.x;
    const double w[3] = {1.0, 1.0, -2.0};   // xx + yy - 2*xy
    double s = 0.0;
    for (int p = 0; p < 3; ++p)
        for (int i = tid; i < TILES_PER_PAIR; i += 256)
            s += w[p] * (double)partials[p * TILES_PER_PAIR + i];
    red[tid] = s;
    __syncthreads();
    for (int st = 128; st > 0; st >>= 1) {
        if (tid < st) red[tid] += red[tid + st];
        __syncthreads();
    }
    if (tid == 0)
        out[0] = (float)(red[0] / ((double)NROWS * (double)NROWS * 5.0));
}

extern "C" void kernel_launch(void* const* d_in, const int* in_sizes, int n_in,
                              void* d_out, int out_size, void* d_ws, size_t ws_size,
                              hipStream_t stream) {
    (void)in_sizes; (void)n_in; (void)out_size; (void)ws_size;
    const float* s = (const float*)d_in[0];
    const float* t = (const float*)d_in[1];
    float* partials = (float*)d_ws;               // 3 * 4096 floats = 48 KB scratch

    dim3 grid(NROWS / BM, NROWS / BN);
    dim3 block(256);
    mmd_tile_kernel<<<grid, block, 0, stream>>>(s, s, partials + 0 * TILES_PER_PAIR, 1); // k_xx
    mmd_tile_kernel<<<grid, block, 0, stream>>>(t, t, partials + 1 * TILES_PER_PAIR, 1); // k_yy
    mmd_tile_kernel<<<grid, block, 0, stream>>>(s, t, partials + 2 * TILES_PER_PAIR, 0); // k_xy
    mmd_finalize<<<1, block, 0, stream>>>(partials, (float*)d_out);
}